// HyperGCN_17171279249556
// MI455X (gfx1250) — compile-verified
//
#include <hip/hip_runtime.h>
#include <hip/hip_bf16.h>

#define DIM 128
#define WPAD 8                    // LDS row padding (floats) to avoid bank conflicts
#define WSTRIDE (DIM + WPAD)      // 136
#define NEG_SLOPE 0.01f
#define N_EDGES_CONST 50000

typedef float v2f __attribute__((ext_vector_type(2)));
typedef float v8f __attribute__((ext_vector_type(8)));

__device__ __forceinline__ void atomAddF(float* p, float v) {
  __hip_atomic_fetch_add(p, v, __ATOMIC_RELAXED, __HIP_MEMORY_SCOPE_AGENT);
}

// ---------------------------------------------------------------------------
// Y[M,128] = X[M,128] @ W[128,128], fp32, via V_WMMA_F32_16X16X4_F32.
// One wave per 16-row strip; 8 column tiles of 16; K swept in steps of 4.
// A frag layout (32b 16x4): lane L (m = L&15), VGPR j -> K = j + 2*(L>>4)
// B frag layout (32b 4x16): lane L (n = L&15), VGPR j -> K = j + 2*(L>>4)
// C/D layout  (32b 16x16): lane L (n = L&15), VGPR r -> M = r + 8*(L>>4)
// ---------------------------------------------------------------------------
__global__ __launch_bounds__(256) void gemm128_wmma(const float* __restrict__ X,
                                                    const float* __restrict__ W,
                                                    float* __restrict__ Y, int M) {
  __shared__ float Ws[DIM * WSTRIDE];
  for (int i = threadIdx.x; i < DIM * DIM; i += 256)
    Ws[(i >> 7) * WSTRIDE + (i & 127)] = W[i];
  __syncthreads();

  const int wave = threadIdx.x >> 5;
  const int lane = threadIdx.x & 31;
  const int tile = blockIdx.x * 8 + wave;
  if (tile * 16 >= M) return;           // wave-uniform; EXEC all-ones in live waves

  const int m0   = tile * 16;
  const int half = lane >> 4;           // 0 or 1
  const int l16  = lane & 15;

  v8f acc[8] = {};
  const float* xrow = X + (size_t)(m0 + l16) * DIM;

  for (int kt = 0; kt < 32; ++kt) {
    const int kb = kt * 4 + half * 2;   // this lane-half's K pair
    v2f a;
    a.x = xrow[kb];
    a.y = xrow[kb + 1];
#pragma unroll
    for (int nt = 0; nt < 8; ++nt) {
      const int n = nt * 16 + l16;
      v2f b;
      b.x = Ws[(kb + 0) * WSTRIDE + n];
      b.y = Ws[(kb + 1) * WSTRIDE + n];
      acc[nt] = __builtin_amdgcn_wmma_f32_16x16x4_f32(
          /*neg_a=*/false, a, /*neg_b=*/false, b,
          /*c_mod=*/(short)0, acc[nt], /*reuse_a=*/false, /*reuse_b=*/false);
    }
  }

  float* yout = Y + (size_t)m0 * DIM;
#pragma unroll
  for (int nt = 0; nt < 8; ++nt) {
    const int n = nt * 16 + l16;
#pragma unroll
    for (int r = 0; r < 8; ++r)
      yout[(size_t)(r + 8 * half) * DIM + n] = acc[nt][r];
  }
}

// ---------------------------------------------------------------------------
// Degree accumulation: D[node] += 1, B[edge] += 1 over all incidences.
// ---------------------------------------------------------------------------
__global__ void degrees_kernel(const long long* __restrict__ nidx,
                               const long long* __restrict__ eidx,
                               float* __restrict__ D, float* __restrict__ B,
                               long long nnz) {
  long long i = (long long)blockIdx.x * blockDim.x + threadIdx.x;
  if (i >= nnz) return;
  atomAddF(&D[nidx[i]], 1.0f);
  atomAddF(&B[eidx[i]], 1.0f);
}

__global__ void invert_kernel(float* __restrict__ D, int nd,
                              float* __restrict__ B, int ne) {
  int i = blockIdx.x * blockDim.x + threadIdx.x;
  if (i < nd) { float d = D[i]; D[i] = (d > 0.0f) ? 1.0f / d : 0.0f; }
  if (i < ne) { float b = B[i]; B[i] = (b > 0.0f) ? 1.0f / b : 0.0f; }
}

// ---------------------------------------------------------------------------
// node -> edge: edge[e,:] += xw[n,:].  32 lanes x float4 cover the 128 dims.
// ---------------------------------------------------------------------------
__global__ void scatter_n2e(const float* __restrict__ xw,
                            const long long* __restrict__ nidx,
                            const long long* __restrict__ eidx,
                            float* __restrict__ edge, long long nnz) {
  long long t = (long long)blockIdx.x * blockDim.x + threadIdx.x;
  long long i = t >> 5;
  if (i >= nnz) return;
  const int c = (int)(t & 31) * 4;
  const long long n = nidx[i];
  const long long e = eidx[i];
  const float4 v = *(const float4*)(xw + (size_t)n * DIM + c);
  float* dst = edge + (size_t)e * DIM + c;
  atomAddF(dst + 0, v.x);
  atomAddF(dst + 1, v.y);
  atomAddF(dst + 2, v.z);
  atomAddF(dst + 3, v.w);
}

// ---------------------------------------------------------------------------
// edge -> node with B^{-1} folded in: node[n,:] += Binv[e] * edge[e,:]
// ---------------------------------------------------------------------------
__global__ void scatter_e2n(const float* __restrict__ edge,
                            const float* __restrict__ Binv,
                            const long long* __restrict__ nidx,
                            const long long* __restrict__ eidx,
                            float* __restrict__ node, long long nnz) {
  long long t = (long long)blockIdx.x * blockDim.x + threadIdx.x;
  long long i = t >> 5;
  if (i >= nnz) return;
  const int c = (int)(t & 31) * 4;
  const long long n = nidx[i];
  const long long e = eidx[i];
  const float s = Binv[e];
  float4 v = *(const float4*)(edge + (size_t)e * DIM + c);
  float* dst = node + (size_t)n * DIM + c;
  atomAddF(dst + 0, s * v.x);
  atomAddF(dst + 1, s * v.y);
  atomAddF(dst + 2, s * v.z);
  atomAddF(dst + 3, s * v.w);
}

// ---------------------------------------------------------------------------
// out = leaky_relu(Dinv[n] * acc + bias)
// ---------------------------------------------------------------------------
__global__ void finalize_kernel(const float* __restrict__ acc,
                                const float* __restrict__ Dinv,
                                const float* __restrict__ bias,
                                float* __restrict__ out, int M) {
  int t = blockIdx.x * blockDim.x + threadIdx.x;
  const int total = M * (DIM / 4);
  if (t >= total) return;
  const int n = t / (DIM / 4);
  const int c = (t % (DIM / 4)) * 4;
  const float dinv = Dinv[n];
  float4 v = *(const float4*)(acc + (size_t)n * DIM + c);
  const float4 bv = *(const float4*)(bias + c);
  float4 r;
  r.x = dinv * v.x + bv.x;
  r.y = dinv * v.y + bv.y;
  r.z = dinv * v.z + bv.z;
  r.w = dinv * v.w + bv.w;
  r.x = (r.x > 0.0f) ? r.x : r.x * NEG_SLOPE;
  r.y = (r.y > 0.0f) ? r.y : r.y * NEG_SLOPE;
  r.z = (r.z > 0.0f) ? r.z : r.z * NEG_SLOPE;
  r.w = (r.w > 0.0f) ? r.w : r.w * NEG_SLOPE;
  *(float4*)(out + (size_t)n * DIM + c) = r;
}

extern "C" void kernel_launch(void* const* d_in, const int* in_sizes, int n_in,
                              void* d_out, int out_size, void* d_ws, size_t ws_size,
                              hipStream_t stream) {
  const float*     X   = (const float*)d_in[0];
  const long long* he  = (const long long*)d_in[1];   // int64 per reference
  const float*     W1  = (const float*)d_in[2];
  const float*     b1  = (const float*)d_in[3];
  const float*     W2  = (const float*)d_in[4];
  const float*     b2  = (const float*)d_in[5];
  float*           out = (float*)d_out;

  const int       M   = in_sizes[0] / DIM;        // 100000 nodes
  const long long NNZ = in_sizes[1] / 2;          // 1.6M incidences
  const int       NE  = N_EDGES_CONST;            // 50000 hyperedges

  const long long* nidx = he;
  const long long* eidx = he + NNZ;

  // Carve workspace
  char* ws = (char*)d_ws;
  size_t off = 0;
  auto carve = [&](size_t bytes) -> void* {
    void* p = ws + off;
    off = (off + bytes + 255) & ~(size_t)255;
    return p;
  };
  float* Dinv = (float*)carve((size_t)M * sizeof(float));
  float* Binv = (float*)carve((size_t)NE * sizeof(float));
  float* bufA = (float*)carve((size_t)M * DIM * sizeof(float));   // xw / x2 / accum2
  float* bufC = (float*)carve((size_t)M * DIM * sizeof(float));   // accum1 / xw2
  float* bufE = (float*)carve((size_t)NE * DIM * sizeof(float));  // edge features

  const size_t nodeBytes = (size_t)M * DIM * sizeof(float);
  const size_t edgeBytes = (size_t)NE * DIM * sizeof(float);

  const int tiles      = (M + 15) / 16;
  const int gemmBlocks = (tiles + 7) / 8;
  const long long nnzT  = NNZ;
  const int degBlocks   = (int)((nnzT + 255) / 256);
  const long long scT   = nnzT * 32;
  const int scBlocks    = (int)((scT + 255) / 256);
  const int invBlocks   = (M + 255) / 256;
  const int finBlocks   = (M * (DIM / 4) + 255) / 256;

  // ---- degrees & inverses --------------------------------------------------
  hipMemsetAsync(Dinv, 0, (size_t)M * sizeof(float), stream);
  hipMemsetAsync(Binv, 0, (size_t)NE * sizeof(float), stream);
  degrees_kernel<<<degBlocks, 256, 0, stream>>>(nidx, eidx, Dinv, Binv, NNZ);
  invert_kernel<<<invBlocks, 256, 0, stream>>>(Dinv, M, Binv, NE);

  // ---- layer 1 -------------------------------------------------------------
  gemm128_wmma<<<gemmBlocks, 256, 0, stream>>>(X, W1, bufA, M);
  hipMemsetAsync(bufE, 0, edgeBytes, stream);
  scatter_n2e<<<scBlocks, 256, 0, stream>>>(bufA, nidx, eidx, bufE, NNZ);
  hipMemsetAsync(bufC, 0, nodeBytes, stream);
  scatter_e2n<<<scBlocks, 256, 0, stream>>>(bufE, Binv, nidx, eidx, bufC, NNZ);
  finalize_kernel<<<finBlocks, 256, 0, stream>>>(bufC, Dinv, b1, bufA, M); // x2 -> bufA

  // ---- layer 2 -------------------------------------------------------------
  gemm128_wmma<<<gemmBlocks, 256, 0, stream>>>(bufA, W2, bufC, M);
  hipMemsetAsync(bufE, 0, edgeBytes, stream);
  scatter_n2e<<<scBlocks, 256, 0, stream>>>(bufC, nidx, eidx, bufE, NNZ);
  hipMemsetAsync(bufA, 0, nodeBytes, stream);
  scatter_e2n<<<scBlocks, 256, 0, stream>>>(bufE, Binv, nidx, eidx, bufA, NNZ);
  finalize_kernel<<<finBlocks, 256, 0, stream>>>(bufA, Dinv, b2, out, M);
}